// FPNPredictor_68556267978764
// MI455X (gfx1250) — compile-verified
//
#include <hip/hip_runtime.h>

typedef __attribute__((ext_vector_type(2))) float v2f;
typedef __attribute__((ext_vector_type(8))) float v8f;

// Global-address-space views so weight loads lower to global_load_b64
// (LOADcnt only) instead of flat_load_b64 (LOADcnt + DScnt + aperture check).
typedef __attribute__((address_space(1))) const float gcf;
typedef __attribute__((address_space(1))) const v2f  gv2f;

#define WMMA_F32(a, b, c) \
  __builtin_amdgcn_wmma_f32_16x16x4_f32(false, (a), false, (b), (short)0, (c), false, false)

namespace {
constexpr int N_ROWS = 65536;
constexpr int D      = 1024;
constexpr int S      = 200;
constexpr int NBBOX  = 324;    // 81*4
constexpr int ROWS   = 32;     // rows per block
constexpr int KC     = 64;     // K chunk staged in LDS
constexpr int T_SEM  = 13;     // 13*16 = 208 virtual cols: 0..199 sem, 200 = W_cls
constexpr int T_ALL  = 34;     // 13 sem-side + 21 bbox-side
constexpr int MAXT   = 9;      // uniform tiles/wave (pads 34 -> 36, tiles 34/35 dummy)

// mask over cls columns 0..79 (classes 1..80): IGNORE2-1 plus 79
constexpr unsigned long long MASK_LO =
    (1ULL << 1) | (1ULL << 2) | (1ULL << 16) | (1ULL << 18) | (1ULL << 29) |
    (1ULL << 37) | (1ULL << 47) | (1ULL << 53) | (1ULL << 63);
constexpr unsigned long long MASK_HI =
    (1ULL << 4) | (1ULL << 8) | (1ULL << 15);   // 68, 72, 79
}

#if __has_builtin(__builtin_amdgcn_global_load_async_to_lds_b128)
#define HAVE_ASYNC_LDS 1
typedef int vi4 __attribute__((ext_vector_type(4)));
typedef __attribute__((address_space(1))) vi4 gvi4;   // global 16B vector
typedef __attribute__((address_space(3))) vi4 lvi4;   // LDS    16B vector
#else
#define HAVE_ASYNC_LDS 0
#endif

static __device__ __forceinline__ void wait_async_zero() {
#if HAVE_ASYNC_LDS
#if __has_builtin(__builtin_amdgcn_s_wait_asynccnt)
    __builtin_amdgcn_s_wait_asynccnt(0);
#else
    asm volatile("s_wait_asynccnt 0x0" ::: "memory");
#endif
#endif
}

__global__ __launch_bounds__(256) void fpn_predictor_fused(
    const float* __restrict__ x,      // [N, D]
    const float* __restrict__ W_cls,  // [1, D]
    const float* __restrict__ b_cls,  // [1]
    const float* __restrict__ W_sem,  // [S, D]
    const float* __restrict__ b_sem,  // [S]
    const float* __restrict__ W_bbox, // [324, D]
    const float* __restrict__ b_bbox, // [324]
    const float* __restrict__ SM,     // [80, S]
    float* __restrict__ out_scores,   // [N, 81]
    float* __restrict__ out_bbox)     // [N, 324]
{
    __shared__ float lds_x[ROWS * KC];     // 8 KB: staged x chunk
    __shared__ float lds_sem[ROWS * S];    // 25.6 KB: sem tile for phase 2

    const int tid     = threadIdx.x;
    const int wave    = tid >> 5;
    const int lane    = tid & 31;
    const int rowHalf = wave >> 2;         // 0/1: which 16-row half
    const int colGrp  = wave & 3;          // tile round-robin group
    const int n       = lane & 15;         // matrix column (B/C/D) / row (A)
    const int kh      = lane >> 4;         // K-half selector for f32 16x16x4
    const int row0    = blockIdx.x * ROWS;

    // ---- per-lane B pointers / biases: ALWAYS valid (clamped), never null ----
    // Out-of-range lanes produce garbage accumulators that are never stored.
    gcf* bptr[MAXT];
    float bias[MAXT];
#pragma unroll
    for (int i = 0; i < MAXT; ++i) {
        const int t = colGrp + 4 * i;            // 0..35 (34,35 dummy)
        if (t < T_SEM) {
            const int v = t * 16 + n;            // 0..207
            if (v < S) { bptr[i] = (gcf*)(W_sem + (size_t)v * D); bias[i] = b_sem[v]; }
            else       { bptr[i] = (gcf*)W_cls;  bias[i] = (v == S) ? b_cls[0] : 0.0f; }
        } else if (t < T_ALL) {
            const int c  = (t - T_SEM) * 16 + n; // 0..335
            const int cc = (c < NBBOX) ? c : (NBBOX - 1);
            bptr[i] = (gcf*)(W_bbox + (size_t)cc * D);
            bias[i] = b_bbox[cc];
        } else {                                  // dummy pad tile
            bptr[i] = (gcf*)(W_sem + (size_t)n * D);
            bias[i] = 0.0f;
        }
    }

    v8f acc[MAXT] = {};

    // ================= Phase 1: [32 x 1024] x {W_sem|W_cls|W_bbox} =================
    for (int k0 = 0; k0 < D; k0 += KC) {
        // stage x[row0..row0+31, k0..k0+63] into LDS
        {
            const int r  = tid >> 3;           // 0..31
            const int kk = (tid & 7) * 8;      // 0..56
            const float* src = x + (size_t)(row0 + r) * D + k0 + kk;
            float* dst = lds_x + r * KC + kk;
#if HAVE_ASYNC_LDS
            __builtin_amdgcn_global_load_async_to_lds_b128(
                (gvi4*)src,       (lvi4*)dst,       0, 0);
            __builtin_amdgcn_global_load_async_to_lds_b128(
                (gvi4*)(src + 4), (lvi4*)(dst + 4), 0, 0);
#else
            const float4 v0 = *(const float4*)(src);
            const float4 v1 = *(const float4*)(src + 4);
            *(float4*)(dst)     = v0;
            *(float4*)(dst + 4) = v1;
#endif
            if (k0 + KC < D)  // prefetch next chunk -> global_prefetch_b8
                __builtin_prefetch(src + KC, 0, 1);
        }
        wait_async_zero();
        __syncthreads();

        const float* aBase = lds_x + (rowHalf * 16 + n) * KC + 2 * kh;
        for (int k4 = 0; k4 < KC; k4 += 4) {
            const v2f a  = *(const v2f*)(aBase + k4);
            const int kg = k0 + k4 + 2 * kh;
            v2f b[MAXT];
#pragma unroll
            for (int i = 0; i < MAXT; ++i)       // 9 unconditional global_load_b64
                b[i] = *(gv2f*)(bptr[i] + kg);
#pragma unroll
            for (int i = 0; i < MAXT; ++i)       // 9 back-to-back WMMAs
                acc[i] = WMMA_F32(a, b[i], acc[i]);
        }
        __syncthreads();
    }

    // ---- scatter phase-1 results: sem -> LDS, b_score + bbox -> HBM ----
#pragma unroll
    for (int i = 0; i < MAXT; ++i) {
        const int t = colGrp + 4 * i;
        if (t < T_SEM) {
            const int v = t * 16 + n;
#pragma unroll
            for (int j = 0; j < 8; ++j) {
                const int   rl  = rowHalf * 16 + j + kh * 8;
                const float val = acc[i][j] + bias[i];
                if (v < S)       lds_sem[rl * S + v] = val;
                else if (v == S) out_scores[(size_t)(row0 + rl) * 81] = val;  // b_score
            }
        } else if (t < T_ALL) {
            const int c = (t - T_SEM) * 16 + n;
            if (c < NBBOX) {
#pragma unroll
                for (int j = 0; j < 8; ++j) {
                    const int rl = rowHalf * 16 + j + kh * 8;
                    out_bbox[(size_t)(row0 + rl) * NBBOX + c] = acc[i][j] + bias[i];
                }
            }
        }
    }
    __syncthreads();

    // ================= Phase 2: cls = (sem @ SM^T) * (1/S), masked =================
    // 5 tiles of 16 cols. Every wave computes tile colGrp AND tile 4 (branch-free
    // inner loop; tile-4 work is redundant on 3 of 4 groups, only colGrp==0 stores).
    v8f acc2[2] = {};
    const float* semA = lds_sem + (rowHalf * 16 + n) * S + 2 * kh;
    gcf* smA = (gcf*)(SM + (size_t)(colGrp * 16 + n) * S + 2 * kh);
    gcf* smB = (gcf*)(SM + (size_t)(64 + n) * S + 2 * kh);
    for (int k = 0; k < S; k += 4) {
        const v2f a  = *(const v2f*)(semA + k);
        const v2f b0 = *(gv2f*)(smA + k);
        const v2f b1 = *(gv2f*)(smB + k);
        acc2[0] = WMMA_F32(a, b0, acc2[0]);
        acc2[1] = WMMA_F32(a, b1, acc2[1]);
    }
#pragma unroll
    for (int ii = 0; ii < 2; ++ii) {
        if (ii == 1 && colGrp != 0) break;       // wave-uniform
        const int  c  = ((ii == 0) ? colGrp : 4) * 16 + n;       // 0..79
        const bool on = (c < 64) ? ((MASK_LO >> c) & 1ULL)
                                 : ((MASK_HI >> (c - 64)) & 1ULL);
        const float scale = on ? (1.0f / (float)S) : 0.0f;
#pragma unroll
        for (int j = 0; j < 8; ++j) {
            const int rl = rowHalf * 16 + j + kh * 8;
            out_scores[(size_t)(row0 + rl) * 81 + 1 + c] = acc2[ii][j] * scale;
        }
    }
}

extern "C" void kernel_launch(void* const* d_in, const int* in_sizes, int n_in,
                              void* d_out, int out_size, void* d_ws, size_t ws_size,
                              hipStream_t stream) {
    const float* x      = (const float*)d_in[0];
    const float* W_cls  = (const float*)d_in[1];
    const float* b_cls  = (const float*)d_in[2];
    const float* W_sem  = (const float*)d_in[3];
    const float* b_sem  = (const float*)d_in[4];
    const float* W_bbox = (const float*)d_in[5];
    const float* b_bbox = (const float*)d_in[6];
    const float* SM     = (const float*)d_in[7];

    float* out_scores = (float*)d_out;                       // [N, 81]
    float* out_bbox   = out_scores + (size_t)N_ROWS * 81;    // [N, 324]

    dim3 grid(N_ROWS / ROWS);   // 2048 blocks
    dim3 block(256);            // 8 waves (wave32)
    fpn_predictor_fused<<<grid, block, 0, stream>>>(
        x, W_cls, b_cls, W_sem, b_sem, W_bbox, b_bbox, SM, out_scores, out_bbox);
}